// CaptionEmbedder_28845000360087
// MI455X (gfx1250) — compile-verified
//
#include <hip/hip_runtime.h>
#include <hip/hip_bf16.h>
#include <stdint.h>

// Problem constants (from reference setup_inputs)
#define VOCAB  32000
#define BDIM   64
#define LDIM   200
#define NENT   50
#define DDIM   512           // floats per row (2048 bytes)
#define WAVES_PER_BLOCK 8    // 256 threads, wave32 -> 8 rows per block

typedef float __attribute__((ext_vector_type(4))) v4f;
typedef int   v4i __attribute__((vector_size(4 * sizeof(int))));

// gfx1250 async global<->LDS DMA builtins (ASYNCcnt-tracked). Guarded so the
// host pass and any toolchain lacking them fall back to plain b128 copies.
#if defined(__has_builtin)
#if __has_builtin(__builtin_amdgcn_global_load_async_to_lds_b128) && \
    __has_builtin(__builtin_amdgcn_global_store_async_from_lds_b128) && \
    __has_builtin(__builtin_amdgcn_s_wait_asynccnt)
#define USE_ASYNC_LDS 1
#endif
#endif

#if defined(USE_ASYNC_LDS)
typedef __attribute__((address_space(1))) v4i* gv4i_p;   // global b128 chunk
typedef __attribute__((address_space(3))) v4i* lv4i_p;   // LDS b128 chunk
#endif

__global__ __launch_bounds__(32 * WAVES_PER_BLOCK)
void caption_embed_gather(const int* __restrict__ caption_indices,   // [B,L]
                          const float* __restrict__ entities,        // [B,NENT,D]
                          const float* __restrict__ words,           // [VOCAB,D]
                          const int* __restrict__ pad_token_p,       // [1]
                          const int* __restrict__ caption_masks,     // [B,L,1]
                          float* __restrict__ out)                   // [B,L,D]
{
    __shared__ __align__(16) float lds[WAVES_PER_BLOCK * DDIM];      // 16 KB

    const int lane = threadIdx.x & 31;
    const int wave = threadIdx.x >> 5;
    const int row  = blockIdx.x * WAVES_PER_BLOCK + wave;            // 0 .. B*L-1
    const int b    = row / LDIM;

    const int ci = caption_indices[row];
    const int m  = caption_masks[row];

    // Select source row (uniform across the wave -> no divergence).
    const float* src;
    if (m == 1) {
        int e = ci - VOCAB;
        if (e < 0 || e >= NENT) e = NENT - 1;
        src = entities + ((size_t)b * NENT + (size_t)e) * DDIM;
    } else {
        int w = (ci >= VOCAB) ? pad_token_p[0] : ci;
        src = words + (size_t)w * DDIM;
    }
    float* dst = out + (size_t)row * DDIM;

#if defined(USE_ASYNC_LDS)
    // memory -> LDS -> memory via gfx1250 async DMA; each wave owns its own
    // 2KB LDS slice, so per-wave s_wait_asynccnt is sufficient (no barrier).
    float* lbase = &lds[wave * DDIM];
    #pragma unroll
    for (int j = 0; j < 4; ++j) {
        const float* g = src + j * 128 + lane * 4;          // 16B per lane
        float*       l = lbase + j * 128 + lane * 4;
        __builtin_amdgcn_global_load_async_to_lds_b128((gv4i_p)g, (lv4i_p)l, 0, 0);
    }
    __builtin_amdgcn_s_wait_asynccnt(0);
    #pragma unroll
    for (int j = 0; j < 4; ++j) {
        float*       g = dst + j * 128 + lane * 4;
        const float* l = lbase + j * 128 + lane * 4;
        __builtin_amdgcn_global_store_async_from_lds_b128((gv4i_p)g, (lv4i_p)l, 0, 0);
    }
    // Outstanding async stores are flushed by S_ENDPGM's implicit wait-idle.
#else
    // Fallback: direct b128 register copy; NT stores keep the 64MB embedding
    // table resident in the 192MB L2 (output is write-once streaming).
    (void)lds;
    #pragma unroll
    for (int j = 0; j < 4; ++j) {
        const v4f v = *(const v4f*)(src + j * 128 + lane * 4);
        __builtin_nontemporal_store(v, (v4f*)(dst + j * 128 + lane * 4));
    }
#endif
}

extern "C" void kernel_launch(void* const* d_in, const int* in_sizes, int n_in,
                              void* d_out, int out_size, void* d_ws, size_t ws_size,
                              hipStream_t stream) {
    (void)in_sizes; (void)n_in; (void)out_size; (void)d_ws; (void)ws_size;
    const int*   caption_indices = (const int*)d_in[0];
    const float* entities        = (const float*)d_in[1];
    const float* words           = (const float*)d_in[2];
    const int*   pad_token_p     = (const int*)d_in[3];
    const int*   caption_masks   = (const int*)d_in[4];
    float*       out             = (float*)d_out;

    const int rows = BDIM * LDIM;                         // 12800
    dim3 grid(rows / WAVES_PER_BLOCK);                    // 1600 blocks
    dim3 block(32 * WAVES_PER_BLOCK);                     // 256 threads (8 waves)
    caption_embed_gather<<<grid, block, 0, stream>>>(
        caption_indices, entities, words, pad_token_p, caption_masks, out);
}